// RGAT_78898549227834
// MI455X (gfx1250) — compile-verified
//
#include <hip/hip_runtime.h>

#define F_IN 768
#define C_OUT 64

typedef float v2f __attribute__((ext_vector_type(2)));
typedef float v8f __attribute__((ext_vector_type(8)));

__device__ __forceinline__ v8f wmma4(v2f a, v2f b, v8f c) {
  // V_WMMA_F32_16X16X4_F32: D = A(16x4,f32) * B(4x16,f32) + C(16x16,f32)
  return __builtin_amdgcn_wmma_f32_16x16x4_f32(
      /*neg_a=*/false, a, /*neg_b=*/false, b,
      /*c_mod=*/(short)0, c, /*reuse_a=*/false, /*reuse_b=*/false);
}

// ---------------------------------------------------------------------------
// v_dst[type][k] = sum_c W_dst[type][k, c] * a_dst[type][c]      (3 x 768)
// ---------------------------------------------------------------------------
__global__ void vdst_kernel(const float* __restrict__ Wwd, const float* __restrict__ awd,
                            const float* __restrict__ Wcd, const float* __restrict__ acd,
                            const float* __restrict__ Wpd, const float* __restrict__ apd,
                            float* __restrict__ v)
{
  const int type = blockIdx.y;
  const int k = blockIdx.x * 256 + threadIdx.x;
  if (k >= F_IN) return;
  const float* W = (type == 0) ? Wwd : (type == 1) ? Wcd : Wpd;
  const float* a = (type == 0) ? awd : (type == 1) ? acd : apd;
  const float* wr = W + (size_t)k * C_OUT;
  float s = 0.f;
  #pragma unroll 8
  for (int c = 0; c < C_OUT; ++c) s += wr[c] * a[c];
  v[type * F_IN + k] = s;
}

// ---------------------------------------------------------------------------
// store one 16x64 output tile + its s_src reduction
// C/D layout: vgpr r -> lanes 0-15: row m0+r, lanes 16-31: row m0+8+r
// ---------------------------------------------------------------------------
__device__ __forceinline__ void store_tile(v8f acc0, v8f acc1, v8f acc2, v8f acc3,
                                           float a0, float a1, float a2, float a3,
                                           float* __restrict__ H, float* __restrict__ ssrc,
                                           int m0, int lane)
{
  const int l15 = lane & 15;
  const int rbase = m0 + ((lane >> 4) << 3);
  #pragma unroll
  for (int r = 0; r < 8; ++r) {
    const int row = rbase + r;
    float* hp = H + (size_t)row * C_OUT + l15;
    hp[0]  = acc0[r];
    hp[16] = acc1[r];
    hp[32] = acc2[r];
    hp[48] = acc3[r];
    float p = acc0[r] * a0 + acc1[r] * a1 + acc2[r] * a2 + acc3[r] * a3;
    p += __shfl_xor(p, 1);
    p += __shfl_xor(p, 2);
    p += __shfl_xor(p, 4);
    p += __shfl_xor(p, 8);
    if (l15 == 0) ssrc[row] = p;
  }
}

// ---------------------------------------------------------------------------
// H = X @ W  (M x 768 @ 768 x 64), plus s_src[m] = H[m,:] . a_src.
// FUSE: also sdst[t*M + m] = X[m,:] . vdst[t*768..]  (3 s_dst matvecs).
// One wave handles TWO 16-row tiles (32 rows) so each B fragment feeds 2 WMMAs.
// A-frag (16x4 f32): lanes 0-15 hold K=k0,k0+1 ; lanes 16-31 hold K=k0+2,k0+3.
// B-frag (4x16 f32): vgpr0 = rows K=k0 / k0+2 , vgpr1 = rows K=k0+1 / k0+3.
// ---------------------------------------------------------------------------
template <bool FUSE>
__global__ void __launch_bounds__(256)
gemm_hs(const float* __restrict__ X, const float* __restrict__ W,
        const float* __restrict__ asrc, float* __restrict__ H,
        float* __restrict__ ssrc, int M,
        const float* __restrict__ vdst, float* __restrict__ sdst)
{
  const int lane = threadIdx.x & 31;
  const int wave = threadIdx.x >> 5;
  const int m0 = (blockIdx.x * 8 + wave) * 32;   // two 16-row tiles per wave
  if (m0 >= M) return;                           // wave-uniform
  const int m1 = m0 + 16;
  const bool tile1_ok = (m1 < M);                // wave-uniform (M % 16 == 0)
  const int l15  = lane & 15;
  const int koff = (lane >> 4) << 1;             // 0 or 2

  const int row0 = m0 + l15;
  int row1 = m1 + l15;
  if (row1 >= M) row1 = M - 1;                   // clamp: safe reads, results discarded
  const float* xp0 = X + (size_t)row0 * F_IN + koff;
  const float* xp1 = X + (size_t)row1 * F_IN + koff;

  v8f c00 = {}, c01 = {}, c02 = {}, c03 = {};
  v8f c10 = {}, c11 = {}, c12 = {}, c13 = {};
  float sw0 = 0.f, sc0 = 0.f, sp0 = 0.f;
  float sw1 = 0.f, sc1 = 0.f, sp1 = 0.f;

  for (int k0 = 0; k0 < F_IN; k0 += 4) {
    v2f av0 = *(const v2f*)(xp0 + k0);
    v2f av1 = *(const v2f*)(xp1 + k0);
    const float* wp = W + (size_t)(k0 + koff) * C_OUT + l15;
    v2f b0 = { wp[0],  wp[C_OUT] };
    v2f b1 = { wp[16], wp[16 + C_OUT] };
    v2f b2 = { wp[32], wp[32 + C_OUT] };
    v2f b3 = { wp[48], wp[48 + C_OUT] };
    c00 = wmma4(av0, b0, c00);
    c10 = wmma4(av1, b0, c10);
    c01 = wmma4(av0, b1, c01);
    c11 = wmma4(av1, b1, c11);
    c02 = wmma4(av0, b2, c02);
    c12 = wmma4(av1, b2, c12);
    c03 = wmma4(av0, b3, c03);
    c13 = wmma4(av1, b3, c13);
    if (FUSE) {
      v2f vw = *(const v2f*)(vdst + (k0 + koff));
      v2f vc = *(const v2f*)(vdst + F_IN + (k0 + koff));
      v2f vp = *(const v2f*)(vdst + 2 * F_IN + (k0 + koff));
      sw0 += av0[0] * vw[0] + av0[1] * vw[1];
      sc0 += av0[0] * vc[0] + av0[1] * vc[1];
      sp0 += av0[0] * vp[0] + av0[1] * vp[1];
      sw1 += av1[0] * vw[0] + av1[1] * vw[1];
      sc1 += av1[0] * vc[0] + av1[1] * vc[1];
      sp1 += av1[0] * vp[0] + av1[1] * vp[1];
    }
  }

  const float a0 = asrc[l15], a1 = asrc[16 + l15], a2 = asrc[32 + l15], a3 = asrc[48 + l15];
  store_tile(c00, c01, c02, c03, a0, a1, a2, a3, H, ssrc, m0, lane);
  if (tile1_ok)
    store_tile(c10, c11, c12, c13, a0, a1, a2, a3, H, ssrc, m1, lane);

  if (FUSE) {
    sw0 += __shfl_xor(sw0, 16);  // combine K%4∈{0,1} half with {2,3} half
    sc0 += __shfl_xor(sc0, 16);
    sp0 += __shfl_xor(sp0, 16);
    sw1 += __shfl_xor(sw1, 16);
    sc1 += __shfl_xor(sc1, 16);
    sp1 += __shfl_xor(sp1, 16);
    if (lane < 16) {
      sdst[m0 + lane] = sw0;
      sdst[(size_t)M + m0 + lane] = sc0;
      sdst[2 * (size_t)M + m0 + lane] = sp0;
      if (tile1_ok) {
        sdst[m1 + lane] = sw1;
        sdst[(size_t)M + m1 + lane] = sc1;
        sdst[2 * (size_t)M + m1 + lane] = sp1;
      }
    }
  }
}

// ---------------------------------------------------------------------------
// init: segment max = -inf, denom = 0; out = bw + bc + bp
// ---------------------------------------------------------------------------
__global__ void init_md(float* __restrict__ m, float* __restrict__ den, int n) {
  int i = blockIdx.x * 256 + threadIdx.x;
  if (i >= n) return;
  m[i] = -INFINITY;
  den[i] = 0.f;
}

__global__ void init_out(float* __restrict__ out, const float* __restrict__ bw,
                         const float* __restrict__ bc, const float* __restrict__ bp, int n) {
  int i = blockIdx.x * 256 + threadIdx.x;
  if (i >= n) return;
  int c = i & (C_OUT - 1);
  out[i] = bw[c] + bc[c] + bp[c];
}

// ---------------------------------------------------------------------------
// segment softmax, 3 passes
// ---------------------------------------------------------------------------
__device__ __forceinline__ void atomicMaxFloat(float* addr, float val) {
  int* ia = (int*)addr;
  int old = *ia;
  while (__int_as_float(old) < val) {
    int assumed = old;
    old = atomicCAS(ia, assumed, __float_as_int(val));
    if (old == assumed) break;
  }
}

__global__ void edge_pass1(const int* __restrict__ src, const int* __restrict__ dst,
                           const float* __restrict__ ssrc, const float* __restrict__ sdst,
                           float* __restrict__ e, float* __restrict__ m, int E)
{
  int i = blockIdx.x * 256 + threadIdx.x;
  if (i >= E) return;
  float v = ssrc[src[i]] + sdst[dst[i]];
  v = (v > 0.f) ? v : 0.2f * v;              // leaky_relu, slope 0.2
  e[i] = v;
  atomicMaxFloat(m + dst[i], v);
}

__global__ void edge_pass2(const int* __restrict__ dst, float* __restrict__ e,
                           const float* __restrict__ m, float* __restrict__ den, int E)
{
  int i = blockIdx.x * 256 + threadIdx.x;
  if (i >= E) return;
  int d = dst[i];
  float w = __expf(e[i] - m[d]);
  e[i] = w;                                   // overwrite score with weight
  atomicAdd(den + d, w);
}

// one wave32 per edge: lane handles channels 2*lane, 2*lane+1
__global__ void __launch_bounds__(256)
edge_pass3(const int* __restrict__ src, const int* __restrict__ dst,
           const float* __restrict__ e, const float* __restrict__ den,
           const float* __restrict__ H, float* __restrict__ out, int E)
{
  const int lane = threadIdx.x & 31;
  const int gw = blockIdx.x * 8 + (threadIdx.x >> 5);
  if (gw >= E) return;
  const int s = src[gw], d = dst[gw];
  const float alpha = e[gw] / (den[d] + 1e-16f);
  const float2 h = ((const float2*)(H + (size_t)s * C_OUT))[lane];
  float* op = out + (size_t)d * C_OUT + lane * 2;
  atomicAdd(op,     alpha * h.x);
  atomicAdd(op + 1, alpha * h.y);
}

// ---------------------------------------------------------------------------
extern "C" void kernel_launch(void* const* d_in, const int* in_sizes, int n_in,
                              void* d_out, int out_size, void* d_ws, size_t ws_size,
                              hipStream_t stream)
{
  const float* x_a = (const float*)d_in[0];
  const float* x_p = (const float*)d_in[1];
  const float* x_v = (const float*)d_in[2];
  const int* w_src = (const int*)d_in[3];
  const int* w_dst = (const int*)d_in[4];
  const int* c_src = (const int*)d_in[5];
  const int* c_dst = (const int*)d_in[6];
  const int* p_src = (const int*)d_in[7];
  const int* p_dst = (const int*)d_in[8];
  const float* Wws = (const float*)d_in[9];
  const float* Wwd = (const float*)d_in[10];
  const float* aws = (const float*)d_in[11];
  const float* awd = (const float*)d_in[12];
  const float* bw  = (const float*)d_in[13];
  const float* Wcs = (const float*)d_in[14];
  const float* Wcd = (const float*)d_in[15];
  const float* acs = (const float*)d_in[16];
  const float* acd = (const float*)d_in[17];
  const float* bc  = (const float*)d_in[18];
  const float* Wps = (const float*)d_in[19];
  const float* Wpd = (const float*)d_in[20];
  const float* aps = (const float*)d_in[21];
  const float* apd = (const float*)d_in[22];
  const float* bp  = (const float*)d_in[23];

  const int Na = in_sizes[0] / F_IN;
  const int Np = in_sizes[1] / F_IN;
  const int Nv = in_sizes[2] / F_IN;
  const int Ew = in_sizes[3];
  const int Ec = in_sizes[5];
  const int Ep = in_sizes[7];
  float* out = (float*)d_out;

  // workspace layout (floats)
  float* q = (float*)d_ws;
  float* vdst = q;  q += 3 * F_IN;
  float* Ha   = q;  q += (size_t)Na * C_OUT;
  float* Hpc  = q;  q += (size_t)Np * C_OUT;
  float* Hv   = q;  q += (size_t)Nv * C_OUT;
  float* s_a  = q;  q += Na;
  float* s_pc = q;  q += Np;
  float* s_v  = q;  q += Nv;
  float* sdst = q;  q += 3 * (size_t)Np;   // [w | c | p]
  float* m    = q;  q += 3 * (size_t)Np;
  float* den  = q;  q += 3 * (size_t)Np;
  float* e_w  = q;  q += Ew;
  float* e_c  = q;  q += Ec;
  float* e_p  = q;  q += Ep;

  // 1) v_dst = W_dst @ a_dst for the 3 edge types
  vdst_kernel<<<dim3((F_IN + 255) / 256, 3), 256, 0, stream>>>(Wwd, awd, Wcd, acd, Wpd, apd, vdst);

  // 2) h_src GEMMs (+ s_src); cites GEMM also fuses the 3 s_dst matvecs over x_paper
  auto tiles = [](int M) { return ((M + 31) / 32 + 7) / 8; };
  gemm_hs<false><<<tiles(Na), 256, 0, stream>>>(x_a, Wws, aws, Ha,  s_a,  Na, nullptr, nullptr);
  gemm_hs<true ><<<tiles(Np), 256, 0, stream>>>(x_p, Wcs, acs, Hpc, s_pc, Np, vdst,    sdst);
  gemm_hs<false><<<tiles(Nv), 256, 0, stream>>>(x_v, Wps, aps, Hv,  s_v,  Nv, nullptr, nullptr);

  // 3) init segment-max / denom / output(bias)
  init_md<<<(3 * Np + 255) / 256, 256, 0, stream>>>(m, den, 3 * Np);
  init_out<<<(Np * C_OUT + 255) / 256, 256, 0, stream>>>(out, bw, bc, bp, Np * C_OUT);

  // 4) pass 1: edge scores + segment max
  edge_pass1<<<(Ew + 255) / 256, 256, 0, stream>>>(w_src, w_dst, s_a,  sdst,          e_w, m,          Ew);
  edge_pass1<<<(Ec + 255) / 256, 256, 0, stream>>>(c_src, c_dst, s_pc, sdst + Np,     e_c, m + Np,     Ec);
  edge_pass1<<<(Ep + 255) / 256, 256, 0, stream>>>(p_src, p_dst, s_v,  sdst + 2 * Np, e_p, m + 2 * Np, Ep);

  // 5) pass 2: exp + segment sum
  edge_pass2<<<(Ew + 255) / 256, 256, 0, stream>>>(w_dst, e_w, m,          den,          Ew);
  edge_pass2<<<(Ec + 255) / 256, 256, 0, stream>>>(c_dst, e_c, m + Np,     den + Np,     Ec);
  edge_pass2<<<(Ep + 255) / 256, 256, 0, stream>>>(p_dst, e_p, m + 2 * Np, den + 2 * Np, Ep);

  // 6) pass 3: weighted scatter of h_src into out
  edge_pass3<<<(Ew + 7) / 8, 256, 0, stream>>>(w_src, w_dst, e_w, den,          Ha,  out, Ew);
  edge_pass3<<<(Ec + 7) / 8, 256, 0, stream>>>(c_src, c_dst, e_c, den + Np,     Hpc, out, Ec);
  edge_pass3<<<(Ep + 7) / 8, 256, 0, stream>>>(p_src, p_dst, e_p, den + 2 * Np, Hv,  out, Ep);
}